// NGramsEmbedding_35588099015425
// MI455X (gfx1250) — compile-verified
//
#include <hip/hip_runtime.h>

// NGramsEmbedding: out[t, :] = sum_{g=0..2} weight[idx[g, t], :]
//   idx:    [3, n_tokens] int32 (n_tokens = 512*128 = 65536)
//   weight: [50257, 256] fp32  (49 MiB -> L2-resident on MI455X's 192 MB L2)
//   out:    [n_tokens, 256] fp32 (64 MiB, streamed once -> non-temporal stores)
//
// Memory-bound kernel: ~113 MiB HBM traffic -> ~5 us floor at 23.3 TB/s.
// 64 threads per token, one float4 per thread: each wave32 issues contiguous
// 512 B global_load_b128 bursts from the weight table and a coalesced NT
// b128 store of the result. Block-wide gather indices are staged into LDS
// once via the gfx1250 async-to-LDS path (ASYNCcnt) instead of 64 redundant
// per-lane loads.

typedef float v4f __attribute__((ext_vector_type(4)));

typedef __attribute__((address_space(1))) int gint_t;  // global (AS1)
typedef __attribute__((address_space(3))) int lint_t;  // LDS    (AS3)

#ifndef __has_builtin
#define __has_builtin(x) 0
#endif

#if __has_builtin(__builtin_amdgcn_global_load_async_to_lds_b32) && \
    __has_builtin(__builtin_amdgcn_s_wait_asynccnt)
#define USE_ASYNC_LDS 1
#else
#define USE_ASYNC_LDS 0
#endif

#define NGRAMS 3
#define TOKENS_PER_BLOCK 4     // 256 threads = 4 tokens * 64 threads/token
#define CHUNKS_PER_TOKEN 64    // 256 floats / 4 per float4

__global__ void __launch_bounds__(256)
ngram_embed_sum_kernel(const int* __restrict__ idx,
                       const float* __restrict__ weight,
                       float* __restrict__ out,
                       int n_tokens)
{
    __shared__ int s_idx[TOKENS_PER_BLOCK * NGRAMS];

    const int tid = threadIdx.x;
    const int block_token0 = blockIdx.x * TOKENS_PER_BLOCK;

    // ---- Stage this block's 12 gather indices into LDS (once per block) ----
    if (tid < TOKENS_PER_BLOCK * NGRAMS) {
        const int t_loc = tid % TOKENS_PER_BLOCK;       // token within block
        const int g     = tid / TOKENS_PER_BLOCK;       // n-gram id
        const int tok   = block_token0 + t_loc;
        if (tok < n_tokens) {
            const int* src = idx + (size_t)g * (size_t)n_tokens + (size_t)tok;
#if USE_ASYNC_LDS
            // gfx1250 async copy: GLOBAL_LOAD_ASYNC_TO_LDS_B32 (ASYNCcnt)
            __builtin_amdgcn_global_load_async_to_lds_b32(
                (gint_t*)src, (lint_t*)&s_idx[tid], 0, 0);
#else
            s_idx[tid] = *src;
#endif
        } else {
            s_idx[tid] = 0;
        }
    }
#if USE_ASYNC_LDS
    __builtin_amdgcn_s_wait_asynccnt(0);
#endif
    __syncthreads();

    // ---- Gather + sum: one float4 chunk per thread ----
    const int t_loc = tid >> 6;          // 0..3  (token within block)
    const int chunk = tid & 63;          // 0..63 (float4 chunk within row)
    const int token = block_token0 + t_loc;
    if (token >= n_tokens) return;

    const int i0 = s_idx[0 * TOKENS_PER_BLOCK + t_loc];
    const int i1 = s_idx[1 * TOKENS_PER_BLOCK + t_loc];
    const int i2 = s_idx[2 * TOKENS_PER_BLOCK + t_loc];

    const v4f* __restrict__ w4 = (const v4f*)weight;

    const v4f a = w4[(size_t)i0 * CHUNKS_PER_TOKEN + chunk];
    const v4f b = w4[(size_t)i1 * CHUNKS_PER_TOKEN + chunk];
    const v4f c = w4[(size_t)i2 * CHUNKS_PER_TOKEN + chunk];
    const v4f r = a + b + c;

    // Output is write-once streaming data: non-temporal store keeps the
    // 64 MiB result from displacing the L2-resident embedding table.
    v4f* o4 = (v4f*)out + (size_t)token * CHUNKS_PER_TOKEN + chunk;
    __builtin_nontemporal_store(r, o4);
}

extern "C" void kernel_launch(void* const* d_in, const int* in_sizes, int n_in,
                              void* d_out, int out_size, void* d_ws, size_t ws_size,
                              hipStream_t stream) {
    const int*   idx    = (const int*)d_in[0];    // [3, seq, batch] int32
    const float* weight = (const float*)d_in[1];  // [50257, 256] fp32
    float*       out    = (float*)d_out;          // [seq, batch, 256] fp32

    const int n_tokens = in_sizes[0] / NGRAMS;    // 512*128 = 65536
    const int blocks = (n_tokens + TOKENS_PER_BLOCK - 1) / TOKENS_PER_BLOCK;

    ngram_embed_sum_kernel<<<dim3(blocks), dim3(256), 0, stream>>>(
        idx, weight, out, n_tokens);
}